// simclr_79637283602621
// MI455X (gfx1250) — compile-verified
//
#include <hip/hip_runtime.h>
#include <math.h>

#define B_ROWS 2048
#define N_ROWS 4096
#define DDIM 256
#define PPROTO 100
#define KNEG 30
#define INV_TEMP 5.0f
#define TEMPF 0.2f

typedef float v2f __attribute__((ext_vector_type(2)));
typedef float v8f __attribute__((ext_vector_type(8)));

// ---------------- row L2 normalize (one block per row, 256 threads) --------
__global__ __launch_bounds__(256) void norm_rows_k(const float* __restrict__ in,
                                                   float* __restrict__ out) {
  int r = blockIdx.x, t = threadIdx.x;
  float v = in[r * DDIM + t];
  __shared__ float red[256];
  red[t] = v * v;
  __syncthreads();
  for (int s = 128; s > 0; s >>= 1) {
    if (t < s) red[t] += red[t + s];
    __syncthreads();
  }
  float inv = 1.0f / fmaxf(sqrtf(red[0]), 1e-12f);
  out[r * DDIM + t] = v * inv;
}

// ---------------- positive-pair dots: posPart[r] = out1_r . out2_r ---------
__global__ __launch_bounds__(256) void pos_dot_k(const float* __restrict__ outN,
                                                 float* __restrict__ posPart) {
  int r = blockIdx.x, t = threadIdx.x;
  float v = outN[r * DDIM + t] * outN[(B_ROWS + r) * DDIM + t];
  __shared__ float red[256];
  red[t] = v;
  __syncthreads();
  for (int s = 128; s > 0; s >>= 1) {
    if (t < s) red[t] += red[t + s];
    __syncthreads();
  }
  if (t == 0) posPart[r] = red[0];
}

// ---------------- prototype distance table pd = w w^T (100x100) -----------
__global__ __launch_bounds__(256) void proto_dist_k(const float* __restrict__ w,
                                                    float* __restrict__ pd) {
  int p = blockIdx.x, t = threadIdx.x;
  __shared__ float wp[DDIM];
  wp[t] = w[p * DDIM + t];
  __syncthreads();
  if (t < PPROTO) {
    float s = 0.f;
    for (int d = 0; d < DDIM; ++d) s += wp[d] * w[t * DDIM + d];
    pd[p * PPROTO + t] = s;
  }
}

// ---------------- concat hard_q1 | hard_q2 ---------------------------------
__global__ void concat_q_k(const int* __restrict__ q1, const int* __restrict__ q2,
                           int* __restrict__ hq) {
  int i = blockIdx.x * 256 + threadIdx.x;
  if (i < B_ROWS) {
    hq[i] = q1[i];
    hq[B_ROWS + i] = q2[i];
  }
}

// ---------------- histogram of hard_q over P bins (single block) -----------
__global__ __launch_bounds__(256) void counts_k(const int* __restrict__ hq,
                                                int* __restrict__ counts) {
  __shared__ int c[PPROTO];
  int t = threadIdx.x;
  if (t < PPROTO) c[t] = 0;
  __syncthreads();
  for (int i = t; i < N_ROWS; i += 256) atomicAdd(&c[hq[i]], 1);
  __syncthreads();
  if (t < PPROTO) counts[t] = c[t];
}

// ---------------- 128-bit membership bitmask per row -----------------------
__global__ void member_k(const int* __restrict__ nq1, const int* __restrict__ nq2,
                         unsigned* __restrict__ member) {
  int i = blockIdx.x * 256 + threadIdx.x;
  if (i >= N_ROWS) return;
  const int* nq = (i < B_ROWS) ? (nq1 + i * KNEG) : (nq2 + (i - B_ROWS) * KNEG);
  unsigned m0 = 0, m1 = 0, m2 = 0, m3 = 0;
  for (int k = 0; k < KNEG; ++k) {
    int t = nq[k];
    unsigned bit = 1u << (t & 31);
    int w = t >> 5;
    if (w == 0) m0 |= bit;
    else if (w == 1) m1 |= bit;
    else if (w == 2) m2 |= bit;
    else m3 |= bit;
  }
  member[4 * i + 0] = m0;
  member[4 * i + 1] = m1;
  member[4 * i + 2] = m2;
  member[4 * i + 3] = m3;
}

// ---------------- per-row stats {rmin, 1/range, mu, 1/(2 var)} -------------
__global__ void stats_k(const int* __restrict__ hq, const int* __restrict__ counts,
                        const float* __restrict__ pd, float4* __restrict__ stats) {
  int i = blockIdx.x * 256 + threadIdx.x;
  if (i >= N_ROWS) return;
  int p = hq[i];
  const float* row = pd + p * PPROTO;
  float minv = 1e30f, maxv = -1e30f, s1 = 0.f, s2 = 0.f;
  for (int t = 0; t < PPROTO; ++t) {
    float ct = (float)(counts[t] - (t == p ? 1 : 0));
    if (ct > 0.f) {
      float v = row[t];
      minv = fminf(minv, v);
      maxv = fmaxf(maxv, v);
      s1 += ct * v;
      s2 += ct * v * v;
    }
  }
  float rmin = fminf(0.f, minv);
  float range = maxv - rmin;
  float invR = (range > 0.f) ? 1.f / range : 0.f;
  const float Nf = (float)N_ROWS;
  float sumOff = (s1 - (Nf - 1.f) * rmin) * invR;
  float e2 = (s2 - 2.f * rmin * s1 + (Nf - 1.f) * rmin * rmin) * invR * invR;
  float mu = sumOff / Nf;
  float var = (e2 - Nf * mu * mu) / (Nf - 1.f);  // ddof = 1
  float inv2v = (var > 0.f) ? 1.f / (2.f * var) : 0.f;
  stats[i] = make_float4(rmin, invR, mu, inv2v);
}

// ---------------- fused GEMM (f32 WMMA) + mask/reweight reduction ----------
// Double-buffered async global->LDS staging (GLOBAL_LOAD_ASYNC_TO_LDS_B128,
// tracked by ASYNCcnt) overlapped with WMMA compute.
#define TILE 64
#define KC 32
#define PITCH 36  // dwords; 36 = 4*9, 9 odd -> 16 distinct mult-of-4 banks

__device__ __forceinline__ void async_ld16(unsigned ldsOff, const float* base,
                                           unsigned voff) {
  asm volatile("global_load_async_to_lds_b128 %0, %1, %2"
               :: "v"(ldsOff), "v"(voff), "s"(base)
               : "memory");
}

__global__ __launch_bounds__(256) void sim_tile_k(
    const float* __restrict__ outN, const int* __restrict__ hq,
    const unsigned* __restrict__ member, const float4* __restrict__ stats,
    const float* __restrict__ pd, float* __restrict__ totPart) {
  __shared__ float lA0[TILE * PITCH];
  __shared__ float lB0[TILE * PITCH];
  __shared__ float lA1[TILE * PITCH];
  __shared__ float lB1[TILE * PITCH];
  __shared__ float red[256];

  const int tid = threadIdx.x;
  const int iBase = blockIdx.y * TILE;
  const int jBase = blockIdx.x * TILE;
  const int lane = tid & 31, wav = tid >> 5;
  const int subM = (wav & 3) * 16;        // 4 M-strips of 16
  const int subN0 = (wav >> 2) * 32;      // 2 N-subtiles per wave
  const int l = lane & 15, half = lane >> 4;

  v8f acc0 = {0.f, 0.f, 0.f, 0.f, 0.f, 0.f, 0.f, 0.f};
  v8f acc1 = {0.f, 0.f, 0.f, 0.f, 0.f, 0.f, 0.f, 0.f};

  const int rowA = subM + l;
  const int rowB0 = subN0 + l;
  const int rowB1 = subN0 + 16 + l;
  const int koff = 2 * half;  // lanes 16-31 hold K+2,K+3

  // Loader mapping: 64x32 floats = 512 float4 per tile; each thread owns 2.
  const int ldRow = tid >> 3;          // 0..31 (second slice: +32)
  const int ldC4 = tid & 7;            // float4 column within 32-wide chunk
  const unsigned ldsRel0 = (unsigned)(ldRow * PITCH * 4 + ldC4 * 16);
  const unsigned ldsRel1 = ldsRel0 + 32u * PITCH * 4;
  const unsigned voffA0 = (unsigned)((iBase + ldRow) * (DDIM * 4) + ldC4 * 16);
  const unsigned voffA1 = voffA0 + 32u * (DDIM * 4);
  const unsigned voffB0 = (unsigned)((jBase + ldRow) * (DDIM * 4) + ldC4 * 16);
  const unsigned voffB1 = voffB0 + 32u * (DDIM * 4);
  const unsigned baseA[2] = {(unsigned)(size_t)lA0, (unsigned)(size_t)lA1};
  const unsigned baseB[2] = {(unsigned)(size_t)lB0, (unsigned)(size_t)lB1};

  // Prologue: stream chunk 0 into buffer 0.
  {
    unsigned kcb = 0;
    async_ld16(baseA[0] + ldsRel0, outN, voffA0 + kcb);
    async_ld16(baseA[0] + ldsRel1, outN, voffA1 + kcb);
    async_ld16(baseB[0] + ldsRel0, outN, voffB0 + kcb);
    async_ld16(baseB[0] + ldsRel1, outN, voffB1 + kcb);
  }

  const int nChunks = DDIM / KC;  // 8
  for (int c = 0; c < nChunks; ++c) {
    const int cur = c & 1;
    if (c + 1 < nChunks) {
      // Stream next chunk into the other buffer while we compute this one.
      const int nxt = 1 - cur;
      unsigned kcb = (unsigned)((c + 1) * KC * 4);
      async_ld16(baseA[nxt] + ldsRel0, outN, voffA0 + kcb);
      async_ld16(baseA[nxt] + ldsRel1, outN, voffA1 + kcb);
      async_ld16(baseB[nxt] + ldsRel0, outN, voffB0 + kcb);
      async_ld16(baseB[nxt] + ldsRel1, outN, voffB1 + kcb);
      // Async loads complete in order: <=4 outstanding => chunk c landed.
      asm volatile("s_wait_asynccnt 0x4" ::: "memory");
    } else {
      asm volatile("s_wait_asynccnt 0x0" ::: "memory");
    }
    __syncthreads();  // chunk c visible to all waves

    const float* cA = cur ? lA1 : lA0;
    const float* cB = cur ? lB1 : lB0;
#pragma unroll
    for (int k = 0; k < KC; k += 4) {
      v2f a = *(const v2f*)&cA[rowA * PITCH + k + koff];
      v2f b0 = *(const v2f*)&cB[rowB0 * PITCH + k + koff];
      v2f b1 = *(const v2f*)&cB[rowB1 * PITCH + k + koff];
      acc0 = __builtin_amdgcn_wmma_f32_16x16x4_f32(false, a, false, b0,
                                                   (short)0, acc0, false, false);
      acc1 = __builtin_amdgcn_wmma_f32_16x16x4_f32(false, a, false, b1,
                                                   (short)0, acc1, false, false);
    }
    __syncthreads();  // all waves done with buffer `cur` before it is refilled
  }

  // Epilogue: C element (M = r + 8*half, N = l) per VGPR r.
  float contrib = 0.f;
#pragma unroll
  for (int sub = 0; sub < 2; ++sub) {
    int j = jBase + subN0 + sub * 16 + l;
    int qj = hq[j];
    int wq = qj >> 5;
    unsigned bq = 1u << (qj & 31);
#pragma unroll
    for (int r = 0; r < 8; ++r) {
      int i = iBase + subM + (half << 3) + r;
      if (i != j && (member[4 * i + wq] & bq)) {
        float4 s = stats[i];
        float pdv = pd[hq[i] * PPROTO + qj];
        float Rn = (pdv - s.x) * s.y;
        float d = Rn - s.z;
        float dot = (sub == 0) ? acc0[r] : acc1[r];
        contrib += expf(dot * INV_TEMP) * expf(d * d * s.w);
      }
    }
  }

  red[tid] = contrib;
  __syncthreads();
  for (int s = 128; s > 0; s >>= 1) {
    if (tid < s) red[tid] += red[tid + s];
    __syncthreads();
  }
  if (tid == 0) totPart[blockIdx.y * gridDim.x + blockIdx.x] = red[0];
}

// ---------------- final deterministic reduction + loss ---------------------
__global__ __launch_bounds__(256) void finalize_k(const float* __restrict__ totPart, int nTot,
                                                  const float* __restrict__ posPart, int nPos,
                                                  float* __restrict__ out) {
  __shared__ float red[256];
  int t = threadIdx.x;
  float s = 0.f;
  for (int i = t; i < nTot; i += 256) s += totPart[i];
  red[t] = s;
  __syncthreads();
  for (int k = 128; k > 0; k >>= 1) {
    if (t < k) red[t] += red[t + k];
    __syncthreads();
  }
  float total = red[0];
  __syncthreads();
  float p = 0.f;
  for (int i = t; i < nPos; i += 256) p += posPart[i];
  red[t] = p;
  __syncthreads();
  for (int k = 128; k > 0; k >>= 1) {
    if (t < k) red[t] += red[t + k];
    __syncthreads();
  }
  if (t == 0) {
    float posSum = red[0];  // sum of dots over B rows
    out[0] = logf(total) - posSum * (1.0f / (TEMPF * (float)B_ROWS));
  }
}

// ---------------------------------------------------------------------------
extern "C" void kernel_launch(void* const* d_in, const int* in_sizes, int n_in,
                              void* d_out, int out_size, void* d_ws, size_t ws_size,
                              hipStream_t stream) {
  (void)in_sizes; (void)n_in; (void)out_size; (void)ws_size;
  const float* x = (const float*)d_in[0];
  const float* xa = (const float*)d_in[1];
  const float* pw = (const float*)d_in[2];
  const int* hq1 = (const int*)d_in[3];
  const int* hq2 = (const int*)d_in[4];
  const int* nq1 = (const int*)d_in[5];
  const int* nq2 = (const int*)d_in[6];

  char* ws = (char*)d_ws;
  size_t off = 0;
  float* outN = (float*)(ws + off);            off += (size_t)N_ROWS * DDIM * 4;   // 4 MB
  float* wN = (float*)(ws + off);              off += (size_t)PPROTO * DDIM * 4;
  float* pd = (float*)(ws + off);              off += (size_t)PPROTO * PPROTO * 4;
  int* hqAll = (int*)(ws + off);               off += (size_t)N_ROWS * 4;
  int* counts = (int*)(ws + off);              off += 512;
  unsigned* member = (unsigned*)(ws + off);    off += (size_t)N_ROWS * 4 * 4;
  float4* stats = (float4*)(ws + off);         off += (size_t)N_ROWS * 16;
  float* posPart = (float*)(ws + off);         off += (size_t)B_ROWS * 4;
  float* totPart = (float*)(ws + off);         off += (size_t)4096 * 4;

  norm_rows_k<<<B_ROWS, 256, 0, stream>>>(x, outN);
  norm_rows_k<<<B_ROWS, 256, 0, stream>>>(xa, outN + (size_t)B_ROWS * DDIM);
  norm_rows_k<<<PPROTO, 256, 0, stream>>>(pw, wN);
  pos_dot_k<<<B_ROWS, 256, 0, stream>>>(outN, posPart);
  proto_dist_k<<<PPROTO, 256, 0, stream>>>(wN, pd);
  concat_q_k<<<(B_ROWS + 255) / 256, 256, 0, stream>>>(hq1, hq2, hqAll);
  counts_k<<<1, 256, 0, stream>>>(hqAll, counts);
  member_k<<<(N_ROWS + 255) / 256, 256, 0, stream>>>(nq1, nq2, member);
  stats_k<<<(N_ROWS + 255) / 256, 256, 0, stream>>>(hqAll, counts, pd, stats);
  dim3 grid(N_ROWS / TILE, N_ROWS / TILE);
  sim_tile_k<<<grid, 256, 0, stream>>>(outN, hqAll, member, stats, pd, totPart);
  finalize_k<<<1, 256, 0, stream>>>(totPart, (N_ROWS / TILE) * (N_ROWS / TILE),
                                    posPart, B_ROWS, (float*)d_out);
}